// LCRLoss_56375740727730
// MI455X (gfx1250) — compile-verified
//
#include <hip/hip_runtime.h>

// Problem constants (from reference)
#define B_DIM 8192
#define C_DIM 1024
#define LAMBDA_CO 0.02f
#define EPS_F 1e-8f

// GEMM tiling: 64x128 WG tile, 8 waves of 32x32 (2x2 WMMA subtiles each)
#define TILE_I 64
#define TILE_J 128
#define NTI (C_DIM / TILE_I)          // 16
#define NTJ (C_DIM / TILE_J)          // 8
#define KSLICES 8
#define KLEN (B_DIM / KSLICES)        // 1024
#define NQPART (KSLICES * NTI * NTJ)  // 1024
#define NBLK_CE 2048

typedef float v2f __attribute__((ext_vector_type(2)));
typedef float v8f __attribute__((ext_vector_type(8)));

// ---------------------------------------------------------------------------
// 1) Weighted BCE partial sums. 64 MB stream -> ~3 us floor at 23.3 TB/s.
//    float4/int4 loads to minimize VMEM instruction count.
// ---------------------------------------------------------------------------
__global__ __launch_bounds__(256) void ce_kernel(const float4* __restrict__ logits4,
                                                 const int4* __restrict__ targets4,
                                                 const float4* __restrict__ priors4,
                                                 float* __restrict__ cepart) {
    const int tid = threadIdx.x;
    float local = 0.f;
    const int total4 = (B_DIM * C_DIM) / 4;
    for (int idx = blockIdx.x * blockDim.x + tid; idx < total4;
         idx += gridDim.x * blockDim.x) {
        const int c4 = idx & ((C_DIM / 4) - 1);     // 16B-aligned column group
        const float4 x = logits4[idx];
        const int4 ti = targets4[idx];
        const float4 pr = priors4[c4];
        const float xs[4] = {x.x, x.y, x.z, x.w};
        const float ts[4] = {(float)ti.x, (float)ti.y, (float)ti.z, (float)ti.w};
        const float ps[4] = {pr.x, pr.y, pr.z, pr.w};
#pragma unroll
        for (int e = 0; e < 4; ++e) {
            const float pw = 1.f / ps[e];
            const float nw = 1.f / (1.f - ps[e]);
            const float p = 1.f / (1.f + expf(-xs[e]));
            local -= pw * ts[e] * logf(p + EPS_F)
                   + nw * (1.f - ts[e]) * logf(1.f - p + EPS_F);
        }
    }
    __shared__ float red[256];
    red[tid] = local;
    __syncthreads();
    for (int s = 128; s > 0; s >>= 1) {
        if (tid < s) red[tid] += red[tid + s];
        __syncthreads();
    }
    if (tid == 0) cepart[blockIdx.x] = red[0];
}

// ---------------------------------------------------------------------------
// 2) rc[i] = sum_{j != i} (co[i,j] + co[j,i])  (co is 4 MB -> L2 resident)
// ---------------------------------------------------------------------------
__global__ __launch_bounds__(256) void rc_kernel(const float* __restrict__ co,
                                                 float* __restrict__ rc) {
    const int i = blockIdx.x;
    const int tid = threadIdx.x;
    float local = 0.f;
    for (int j = tid; j < C_DIM; j += 256) {
        if (j != i) local += co[i * C_DIM + j] + co[j * C_DIM + i];
    }
    __shared__ float red[256];
    red[tid] = local;
    __syncthreads();
    for (int s = 128; s > 0; s >>= 1) {
        if (tid < s) red[tid] += red[tid + s];
        __syncthreads();
    }
    if (tid == 0) rc[i] = red[0];
}

// ---------------------------------------------------------------------------
// 3) Fused Gram (X^T X) via V_WMMA_F32_16X16X4_F32.
//    Each wave: 2x2 accumulator subtiles (32x32), so per k-step of 4:
//    8 b32 loads (2 base pointers + immediate offsets 0/64/4096/4160) feed
//    4 WMMAs. co-weighted tile reduction -> 1 partial per WG; Gram diagonal
//    written per K-slice for the s2 term (conflict-free slots).
// ---------------------------------------------------------------------------
__global__ __launch_bounds__(256) void gram_wmma_kernel(const float* __restrict__ X,
                                                        const float* __restrict__ co,
                                                        float* __restrict__ qpart,
                                                        float* __restrict__ s2part) {
    const int tp = blockIdx.x;            // tile pair 0..NTI*NTJ-1
    const int ks = blockIdx.y;            // K slice
    const int ti = tp / NTJ;
    const int tj = tp - ti * NTJ;
    const int tid = threadIdx.x;
    const int wave = tid >> 5;
    const int lane = tid & 31;
    const int wi = wave >> 2;             // 0..1  (32-row strip)
    const int wj = wave & 3;              // 0..3  (32-col strip)
    const int i0 = ti * TILE_I + wi * 32;
    const int j0 = tj * TILE_J + wj * 32;

    // A/B fragment addressing (ISA 32-bit 16x4 layout):
    // lanes 0-15 = M/N 0..15 with {v0,v1} = rows {K0,K1}; lanes 16-31 = {K2,K3}.
    const int m = lane & 15;
    const int kb = (lane >> 4) << 1;      // 0 or 2

    v8f acc[2][2] = {};
    const float* pI = X + (size_t)(ks * KLEN + kb) * C_DIM + i0 + m;
    const float* pJ = X + (size_t)(ks * KLEN + kb) * C_DIM + j0 + m;

#pragma unroll 2
    for (int k = 0; k < KLEN; k += 4) {
        v2f a0, a1, b0, b1;
        a0.x = pI[0];            // row k+kb,   col i0+m
        a1.x = pI[16];           // row k+kb,   col i0+16+m
        a0.y = pI[C_DIM];        // row k+kb+1
        a1.y = pI[C_DIM + 16];
        b0.x = pJ[0];
        b1.x = pJ[16];
        b0.y = pJ[C_DIM];
        b1.y = pJ[C_DIM + 16];
        acc[0][0] = __builtin_amdgcn_wmma_f32_16x16x4_f32(false, a0, false, b0,
                                                          (short)0, acc[0][0], false, false);
        acc[0][1] = __builtin_amdgcn_wmma_f32_16x16x4_f32(false, a0, false, b1,
                                                          (short)0, acc[0][1], false, false);
        acc[1][0] = __builtin_amdgcn_wmma_f32_16x16x4_f32(false, a1, false, b0,
                                                          (short)0, acc[1][0], false, false);
        acc[1][1] = __builtin_amdgcn_wmma_f32_16x16x4_f32(false, a1, false, b1,
                                                          (short)0, acc[1][1], false, false);
        pI += 4 * C_DIM;
        pJ += 4 * C_DIM;
    }

    // C/D layout: VGPR v, lanes 0-15 -> M=v, N=lane; lanes 16-31 -> M=v+8.
    const int n = lane & 15;
    const int mh = (lane >> 4) << 3;
    float wsum = 0.f;
#pragma unroll
    for (int is = 0; is < 2; ++is) {
#pragma unroll
        for (int js = 0; js < 2; ++js) {
#pragma unroll
            for (int v = 0; v < 8; ++v) {
                const int gi = i0 + is * 16 + v + mh;
                const int gj = j0 + js * 16 + n;
                const float g = acc[is][js][v];
                const float cw = co[gi * C_DIM + gj];           // always load
                wsum += (gi != gj ? cw : 0.f) * g;              // cndmask mask
                if (gi == gj) s2part[ks * C_DIM + gi] = g;      // diag partial
            }
        }
    }

    __shared__ float red[256];
    red[tid] = wsum;
    __syncthreads();
    for (int s = 128; s > 0; s >>= 1) {
        if (tid < s) red[tid] += red[tid + s];
        __syncthreads();
    }
    if (tid == 0) qpart[ks * (NTI * NTJ) + tp] = red[0];
}

// ---------------------------------------------------------------------------
// 4) Final deterministic combine -> scalar loss
// ---------------------------------------------------------------------------
__global__ __launch_bounds__(256) void final_kernel(const float* __restrict__ cepart,
                                                    const float* __restrict__ qpart,
                                                    const float* __restrict__ s2part,
                                                    const float* __restrict__ rc,
                                                    float* __restrict__ out) {
    __shared__ float red[256];
    const int tid = threadIdx.x;

    float ce = 0.f;
    for (int i = tid; i < NBLK_CE; i += 256) ce += cepart[i];
    red[tid] = ce;
    __syncthreads();
    for (int s = 128; s > 0; s >>= 1) {
        if (tid < s) red[tid] += red[tid + s];
        __syncthreads();
    }
    const float ce_sum = red[0];
    __syncthreads();

    float q = 0.f;
    for (int i = tid; i < NQPART; i += 256) q += qpart[i];
    red[tid] = q;
    __syncthreads();
    for (int s = 128; s > 0; s >>= 1) {
        if (tid < s) red[tid] += red[tid + s];
        __syncthreads();
    }
    const float qsum = red[0];   // sum_ij comask * G_ij (unnormalized)
    __syncthreads();

    float s1 = 0.f;
    for (int i = tid; i < C_DIM; i += 256) {
        float g = 0.f;
        for (int k = 0; k < KSLICES; ++k) g += s2part[k * C_DIM + i];
        s1 += (g / (float)B_DIM) * rc[i];   // s2_i * (rowsum_i + colsum_i)
    }
    red[tid] = s1;
    __syncthreads();
    for (int s = 128; s > 0; s >>= 1) {
        if (tid < s) red[tid] += red[tid + s];
        __syncthreads();
    }

    if (tid == 0) {
        const float co_loss = red[0] - 2.f * qsum / (float)B_DIM;
        out[0] = ce_sum / (float)(B_DIM * C_DIM)
               + LAMBDA_CO * co_loss / (float)(C_DIM * C_DIM);
    }
}

// ---------------------------------------------------------------------------
extern "C" void kernel_launch(void* const* d_in, const int* in_sizes, int n_in,
                              void* d_out, int out_size, void* d_ws, size_t ws_size,
                              hipStream_t stream) {
    (void)in_sizes; (void)n_in; (void)out_size; (void)ws_size;
    const float* logits  = (const float*)d_in[0];
    const int*   targets = (const int*)d_in[1];
    const float* co      = (const float*)d_in[2];
    const float* priors  = (const float*)d_in[3];
    float* out = (float*)d_out;

    float* ws = (float*)d_ws;
    float* cepart = ws;                       // NBLK_CE   (2048)
    float* qpart  = cepart + NBLK_CE;         // NQPART    (1024)
    float* s2part = qpart + NQPART;           // KSLICES*C (8192)
    float* rc     = s2part + KSLICES * C_DIM; // C_DIM     (1024)

    ce_kernel<<<NBLK_CE, 256, 0, stream>>>((const float4*)logits, (const int4*)targets,
                                           (const float4*)priors, cepart);
    rc_kernel<<<C_DIM, 256, 0, stream>>>(co, rc);
    gram_wmma_kernel<<<dim3(NTI * NTJ, KSLICES), 256, 0, stream>>>(logits, co, qpart, s2part);
    final_kernel<<<1, 256, 0, stream>>>(cepart, qpart, s2part, rc, out);
}